// SpatialTransformer_5961414606874
// MI455X (gfx1250) — compile-verified
//
#include <hip/hip_runtime.h>

#define BLOCK 256

namespace {
constexpr int Bb = 2, Dd = 160, Hh = 192, Ww = 160;
constexpr int Nvox = Dd * Hh * Ww;        // 4,915,200 voxels per batch
constexpr int TOT  = Bb * Nvox;           // 9,830,400 outputs
constexpr int NTILES = TOT / BLOCK;       // 38,400 tiles of 256 voxels (exact)
constexpr int NBLK = 1920;                // persistent blocks -> 20 tiles each
}

// Builtin signature (probe-confirmed): typed int* pointers, non-const.
#define GPTR(p) ((__attribute__((address_space(1))) int*)(p))
#define LPTR(p) ((__attribute__((address_space(3))) int*)(p))

// One f32 per lane: global -> LDS via the async data path (tracked by ASYNCcnt).
// cpol=1 -> TH_LOAD_NT: flow is a read-once stream; keep it out of L2 so the
// 8-way-gathered src volume stays resident (192 MB L2 vs 118 MB flow stream).
__device__ __forceinline__ void async_f32_nt(const float* g, float* l) {
#if __has_builtin(__builtin_amdgcn_global_load_async_to_lds_b32)
  __builtin_amdgcn_global_load_async_to_lds_b32(GPTR(g), LPTR(l), 0, 1);
#else
  *l = __builtin_nontemporal_load(g);  // fallback: synchronous copy
#endif
}

template <int Ncnt>
__device__ __forceinline__ void wait_async() {
#if __has_builtin(__builtin_amdgcn_s_wait_asynccnt)
  __builtin_amdgcn_s_wait_asynccnt(Ncnt);
#else
  asm volatile("s_wait_asynccnt %0" :: "i"(Ncnt) : "memory");
#endif
}

__global__ __launch_bounds__(BLOCK)
void SpatialTransformer_5961414606874_kernel(const float* __restrict__ src,
                                             const float* __restrict__ flow,
                                             float* __restrict__ out) {
  // Double-buffered flow staging: [buf][channel][lane]  (6 KB LDS)
  __shared__ float sflow[2][3][BLOCK];
  const int tid    = (int)threadIdx.x;
  const int stride = (int)gridDim.x;

  auto stage = [&](int tile, int buf) {
    const int gid = tile * BLOCK + tid;          // global voxel id
    const int b   = gid / Nvox;
    const int n   = gid - b * Nvox;
    const float* fb = flow + b * 3 * Nvox + n;   // [B,3,D,H,W]
    async_f32_nt(fb,            &sflow[buf][0][tid]);
    async_f32_nt(fb + Nvox,     &sflow[buf][1][tid]);
    async_f32_nt(fb + 2 * Nvox, &sflow[buf][2][tid]);
  };

  int t = (int)blockIdx.x;
  if (t >= NTILES) return;
  int buf = 0;
  stage(t, 0);  // prime buffer 0

  for (; t < NTILES; t += stride) {
    const int tn = t + stride;
    if (tn < NTILES) {
      stage(tn, buf ^ 1);   // prefetch next tile's flow while we gather
      wait_async<3>();      // async "done" is in-order: <=3 => current buf ready
    } else {
      wait_async<0>();
    }

    const int gid = t * BLOCK + tid;
    const int b   = gid / Nvox;
    const int n   = gid - b * Nvox;
    const int wx  = n % Ww;
    const int hy  = (n / Ww) % Hh;
    const int dz  = n / (Ww * Hh);

    const float pd = (float)dz + sflow[buf][0][tid];
    const float ph = (float)hy + sflow[buf][1][tid];
    const float pw = (float)wx + sflow[buf][2][tid];

    const float d0f = floorf(pd), h0f = floorf(ph), w0f = floorf(pw);
    const float td = pd - d0f, th = ph - h0f, tw = pw - w0f;
    const int d0 = (int)d0f, h0 = (int)h0f, w0 = (int)w0f;
    const int d1 = d0 + 1,  h1 = h0 + 1,  w1 = w0 + 1;

    // Per-axis weights with the zeros-padding boundary mask folded in.
    const float wd0 = ((unsigned)d0 < (unsigned)Dd ? 1.f : 0.f) * (1.f - td);
    const float wd1 = ((unsigned)d1 < (unsigned)Dd ? 1.f : 0.f) * td;
    const float wh0 = ((unsigned)h0 < (unsigned)Hh ? 1.f : 0.f) * (1.f - th);
    const float wh1 = ((unsigned)h1 < (unsigned)Hh ? 1.f : 0.f) * th;
    const float ww0 = ((unsigned)w0 < (unsigned)Ww ? 1.f : 0.f) * (1.f - tw);
    const float ww1 = ((unsigned)w1 < (unsigned)Ww ? 1.f : 0.f) * tw;

    // Clamped corner indices (reference clips then masks -> identical result).
    const int cd0 = min(max(d0, 0), Dd - 1), cd1 = min(max(d1, 0), Dd - 1);
    const int ch0 = min(max(h0, 0), Hh - 1), ch1 = min(max(h1, 0), Hh - 1);
    const int cw0 = min(max(w0, 0), Ww - 1), cw1 = min(max(w1, 0), Ww - 1);

    const float* sb = src + b * Nvox;
    const int r00 = (cd0 * Hh + ch0) * Ww;
    const int r01 = (cd0 * Hh + ch1) * Ww;
    const int r10 = (cd1 * Hh + ch0) * Ww;
    const int r11 = (cd1 * Hh + ch1) * Ww;

    // 8-way gather; spatially local (flow ~ N(0,1)) -> high L0/L2 hit rate.
    const float v000 = sb[r00 + cw0], v001 = sb[r00 + cw1];
    const float v010 = sb[r01 + cw0], v011 = sb[r01 + cw1];
    const float v100 = sb[r10 + cw0], v101 = sb[r10 + cw1];
    const float v110 = sb[r11 + cw0], v111 = sb[r11 + cw1];

    const float acc =
        wd0 * (wh0 * (ww0 * v000 + ww1 * v001) +
               wh1 * (ww0 * v010 + ww1 * v011)) +
        wd1 * (wh0 * (ww0 * v100 + ww1 * v101) +
               wh1 * (ww0 * v110 + ww1 * v111));

    // Streaming output: NT store keeps the src gather stream resident in L2.
    __builtin_nontemporal_store(acc, out + gid);
    buf ^= 1;
  }
}

extern "C" void kernel_launch(void* const* d_in, const int* in_sizes, int n_in,
                              void* d_out, int out_size, void* d_ws, size_t ws_size,
                              hipStream_t stream) {
  (void)in_sizes; (void)n_in; (void)out_size; (void)d_ws; (void)ws_size;
  const float* src  = (const float*)d_in[0];
  const float* flow = (const float*)d_in[1];
  float* out = (float*)d_out;

  dim3 grid(NBLK < NTILES ? NBLK : NTILES);
  SpatialTransformer_5961414606874_kernel<<<grid, BLOCK, 0, stream>>>(src, flow, out);
}